// HierarchicalGrid_52759378264714
// MI455X (gfx1250) — compile-verified
//
#include <hip/hip_runtime.h>
#include <hip/hip_bf16.h>

#define FEAT 8

typedef float v4f __attribute__((ext_vector_type(4)));

__device__ __forceinline__ v4f lerp4(v4f a, v4f b, float w) {
    return a * (1.0f - w) + b * w;
}

// One level of border-clamped, align_corners-style trilinear sampling,
// faithfully replicating the reference's double-rescale quirk:
//   gc = (w - LO)/SIZE * (res-1);  px = clip((gc+1)*0.5*(res-1), 0, res-1)
template <int RES>
__device__ __forceinline__ void sample_level(const float* __restrict__ g,
                                             float wx_, float wy_, float wz_,
                                             v4f* out_lo, v4f* out_hi) {
    const float rm1 = (float)(RES - 1);

    float gx = (wx_ - (-100.0f)) / 200.0f * rm1;
    float gy = (wy_ - (-100.0f)) / 200.0f * rm1;
    float gz = (wz_ - (-10.0f))  / 60.0f  * rm1;

    float px = fminf(fmaxf((gx + 1.0f) * 0.5f * rm1, 0.0f), rm1);
    float py = fminf(fmaxf((gy + 1.0f) * 0.5f * rm1, 0.0f), rm1);
    float pz = fminf(fmaxf((gz + 1.0f) * 0.5f * rm1, 0.0f), rm1);

    int x0 = (int)floorf(px);
    int y0 = (int)floorf(py);
    int z0 = (int)floorf(pz);
    int x1 = min(x0 + 1, RES - 1);
    int y1 = min(y0 + 1, RES - 1);
    int z1 = min(z0 + 1, RES - 1);

    float wx = px - (float)x0;
    float wy = py - (float)y0;
    float wz = pz - (float)z0;

    // cell index -> v4f index: ((z*RES+y)*RES+x)*FEAT floats = *2 v4f's
    const v4f* g4 = reinterpret_cast<const v4f*>(g);
    int zy00 = (z0 * RES + y0) * RES;
    int zy01 = (z0 * RES + y1) * RES;
    int zy10 = (z1 * RES + y0) * RES;
    int zy11 = (z1 * RES + y1) * RES;

    int i000 = (zy00 + x0) * 2, i001 = (zy00 + x1) * 2;
    int i010 = (zy01 + x0) * 2, i011 = (zy01 + x1) * 2;
    int i100 = (zy10 + x0) * 2, i101 = (zy10 + x1) * 2;
    int i110 = (zy11 + x0) * 2, i111 = (zy11 + x1) * 2;

    // Issue all 16 independent b128 loads up front: deep LOADcnt pipelining,
    // all served from the 192MB L2 (grids total 137MB and stay resident).
    v4f a0 = g4[i000],     a1 = g4[i001],     a2 = g4[i010],     a3 = g4[i011];
    v4f a4 = g4[i100],     a5 = g4[i101],     a6 = g4[i110],     a7 = g4[i111];
    v4f b0 = g4[i000 + 1], b1 = g4[i001 + 1], b2 = g4[i010 + 1], b3 = g4[i011 + 1];
    v4f b4 = g4[i100 + 1], b5 = g4[i101 + 1], b6 = g4[i110 + 1], b7 = g4[i111 + 1];

    v4f c00 = lerp4(a0, a1, wx);
    v4f c01 = lerp4(a2, a3, wx);
    v4f c10 = lerp4(a4, a5, wx);
    v4f c11 = lerp4(a6, a7, wx);
    v4f c0  = lerp4(c00, c01, wy);
    v4f c1  = lerp4(c10, c11, wy);
    *out_lo = lerp4(c0, c1, wz);

    v4f d00 = lerp4(b0, b1, wx);
    v4f d01 = lerp4(b2, b3, wx);
    v4f d10 = lerp4(b4, b5, wx);
    v4f d11 = lerp4(b6, b7, wx);
    v4f d0  = lerp4(d00, d01, wy);
    v4f d1  = lerp4(d10, d11, wy);
    *out_hi = lerp4(d0, d1, wz);
}

__global__ __launch_bounds__(256) void HierarchicalGrid_52759378264714_kernel(
        const float* __restrict__ world,
        const float* __restrict__ g0, const float* __restrict__ g1,
        const float* __restrict__ g2, const float* __restrict__ g3,
        float* __restrict__ out, int N) {
    const int stride = (int)(gridDim.x * blockDim.x);
    for (int n = (int)(blockIdx.x * blockDim.x + threadIdx.x); n < N; n += stride) {
        // Prefetch next iteration's coords (gfx1250 global_prefetch_b8).
        int pn = n + stride;
        if (pn < N) __builtin_prefetch(world + (size_t)pn * 3, 0, 1);

        // Coords are streamed once: non-temporal loads, don't pollute L2.
        float x = __builtin_nontemporal_load(world + (size_t)n * 3 + 0);
        float y = __builtin_nontemporal_load(world + (size_t)n * 3 + 1);
        float z = __builtin_nontemporal_load(world + (size_t)n * 3 + 2);

        float* o = out + (size_t)n * 32;  // 128B-aligned per point
        v4f lo, hi;

        // Output (256MB) is write-once: NT stores keep it from evicting the
        // L2-resident grids (137MB of 192MB L2).
        sample_level<32>(g0, x, y, z, &lo, &hi);
        __builtin_nontemporal_store(lo, (v4f*)(o + 0));
        __builtin_nontemporal_store(hi, (v4f*)(o + 4));

        sample_level<64>(g1, x, y, z, &lo, &hi);
        __builtin_nontemporal_store(lo, (v4f*)(o + 8));
        __builtin_nontemporal_store(hi, (v4f*)(o + 12));

        sample_level<128>(g2, x, y, z, &lo, &hi);
        __builtin_nontemporal_store(lo, (v4f*)(o + 16));
        __builtin_nontemporal_store(hi, (v4f*)(o + 20));

        sample_level<128>(g3, x, y, z, &lo, &hi);
        __builtin_nontemporal_store(lo, (v4f*)(o + 24));
        __builtin_nontemporal_store(hi, (v4f*)(o + 28));
    }
}

extern "C" void kernel_launch(void* const* d_in, const int* in_sizes, int n_in,
                              void* d_out, int out_size, void* d_ws, size_t ws_size,
                              hipStream_t stream) {
    const float* world = (const float*)d_in[0];
    const float* g0    = (const float*)d_in[1];
    const float* g1    = (const float*)d_in[2];
    const float* g2    = (const float*)d_in[3];
    const float* g3    = (const float*)d_in[4];
    float* out = (float*)d_out;

    int N = in_sizes[0] / 3;  // world_coords is [N,3]

    // 4096 blocks x 256 threads = 1M lanes (32K wave32s): enough iterations
    // (~2 points/lane) for the coord prefetch to matter, massive latency hiding.
    int blocks = 4096;
    int maxBlocks = (N + 255) / 256;
    if (blocks > maxBlocks) blocks = maxBlocks;
    if (blocks < 1) blocks = 1;

    HierarchicalGrid_52759378264714_kernel<<<blocks, 256, 0, stream>>>(
        world, g0, g1, g2, g3, out, N);
}